// MatchingLayer_13082470383656
// MI455X (gfx1250) — compile-verified
//
#include <hip/hip_runtime.h>
#include <math.h>

typedef __attribute__((ext_vector_type(2))) float v2f;
typedef __attribute__((ext_vector_type(8))) float v8f;

#define B_    32
#define S_    64
#define H_    300
#define L_    20
#define KT    60          // K tile (300 = 5*60, multiple of 4)
#define KTP   62          // padded LDS row stride (even -> 8B aligned v2f loads)
#define OUTC  160
#define OUTQOFF (B_*S_*OUTC)   // 327680
#define NEG_INF (-3.4e38f)

// ---------------------------------------------------------------------------
// Kernel 1: maxpool strategy.  For each (b, half, l):
//   C[m][n] = sum_h P[m,h]*Q[n,h]*w[h]^2  via V_WMMA_F32_16X16X4_F32,
//   cos = C / (|Pw_m| |Qw_n|);  out_p = rowmax, out_q = colmax.
// ---------------------------------------------------------------------------
__global__ __launch_bounds__(128)
void maxpool_kernel(const float* __restrict__ p, const float* __restrict__ q,
                    const float* __restrict__ W, float* __restrict__ out)
{
    __shared__ float As[S_][KTP];
    __shared__ float Bs[S_][KTP];
    __shared__ float nA[S_];
    __shared__ float nB[S_];
    __shared__ float cosBuf[S_][S_ + 1];

    const int idx  = blockIdx.x;
    const int l    = idx % L_;
    const int hf   = (idx / L_) & 1;     // 0 = fw (W[2]), 1 = bw (W[3])
    const int b    = idx / (2 * L_);
    const int tid  = threadIdx.x;
    const int lane = tid & 31;
    const int wv   = tid >> 5;           // wave id -> M strip
    const int lo   = lane & 15;
    const int hi   = lane >> 4;

    const float* Ph = p + (b * S_) * (2 * H_) + hf * H_;
    const float* Qh = q + (b * S_) * (2 * H_) + hf * H_;
    const float* Wr = W + ((2 + hf) * L_ + l) * H_;

    v8f c0 = {}, c1 = {}, c2 = {}, c3 = {};
    float nacc = 0.f;

    for (int kt = 0; kt < H_; kt += KT) {
        __syncthreads();
        for (int i = tid; i < S_ * KT; i += 128) {
            int s = i / KT, k = i % KT;
            float wval = Wr[kt + k];
            As[s][k] = Ph[s * (2 * H_) + kt + k] * wval;
            Bs[s][k] = Qh[s * (2 * H_) + kt + k] * wval;
        }
        __syncthreads();
        // per-row squared norms (threads 0..63 -> A rows, 64..127 -> B rows)
        if (tid < S_) {
            for (int k = 0; k < KT; ++k) { float v = As[tid][k]; nacc += v * v; }
        } else {
            int r = tid - S_;
            for (int k = 0; k < KT; ++k) { float v = Bs[r][k]; nacc += v * v; }
        }
        // WMMA: A frag lanes 0-15 hold K{0,1}, lanes 16-31 hold K{2,3}; B mirrored.
        const int m = wv * 16 + lo;
        for (int kb = 0; kb < KT; kb += 4) {
            int ka = kb + 2 * hi;
            v2f a  = *(const v2f*)&As[m][ka];
            v2f b0 = *(const v2f*)&Bs[ 0 + lo][ka];
            v2f b1 = *(const v2f*)&Bs[16 + lo][ka];
            v2f b2 = *(const v2f*)&Bs[32 + lo][ka];
            v2f b3 = *(const v2f*)&Bs[48 + lo][ka];
            c0 = __builtin_amdgcn_wmma_f32_16x16x4_f32(false, a, false, b0, (short)0, c0, false, false);
            c1 = __builtin_amdgcn_wmma_f32_16x16x4_f32(false, a, false, b1, (short)0, c1, false, false);
            c2 = __builtin_amdgcn_wmma_f32_16x16x4_f32(false, a, false, b2, (short)0, c2, false, false);
            c3 = __builtin_amdgcn_wmma_f32_16x16x4_f32(false, a, false, b3, (short)0, c3, false, false);
        }
    }
    __syncthreads();
    if (tid < S_) nA[tid] = sqrtf(nacc); else nB[tid - S_] = sqrtf(nacc);
    __syncthreads();

    // C layout: VGPR j, lanes 0-15 -> M=j, lanes 16-31 -> M=j+8; N = lane&15.
    #pragma unroll
    for (int j = 0; j < 8; ++j) {
        int m = wv * 16 + 8 * hi + j;
        float na = nA[m];
        cosBuf[m][ 0 + lo] = c0[j] / (na * nB[ 0 + lo]);
        cosBuf[m][16 + lo] = c1[j] / (na * nB[16 + lo]);
        cosBuf[m][32 + lo] = c2[j] / (na * nB[32 + lo]);
        cosBuf[m][48 + lo] = c3[j] / (na * nB[48 + lo]);
    }
    __syncthreads();

    if (tid < S_) {            // mv_p: max over q (columns)
        float mx = NEG_INF;
        for (int n = 0; n < S_; ++n) mx = fmaxf(mx, cosBuf[tid][n]);
        out[(b * S_ + tid) * OUTC + (2 + hf) * L_ + l] = mx;
    } else {                   // mv_q: max over p (rows)
        int n = tid - S_;
        float mx = NEG_INF;
        for (int m = 0; m < S_; ++m) mx = fmaxf(mx, cosBuf[m][n]);
        out[OUTQOFF + (b * S_ + n) * OUTC + (2 + hf) * L_ + l] = mx;
    }
}

// ---------------------------------------------------------------------------
// Kernel 2: attentive raw cosine matrix per (b, half) via WMMA (unweighted),
// plus row sums (denominator for q_vec mean) and col sums (for p_vec mean).
// ---------------------------------------------------------------------------
__global__ __launch_bounds__(128)
void attn_cos_kernel(const float* __restrict__ p, const float* __restrict__ q,
                     float* __restrict__ cosW, float* __restrict__ rowSum,
                     float* __restrict__ colSum)
{
    __shared__ float As[S_][KTP];
    __shared__ float Bs[S_][KTP];
    __shared__ float nA[S_];
    __shared__ float nB[S_];
    __shared__ float cosBuf[S_][S_ + 1];

    const int hf   = blockIdx.x & 1;
    const int b    = blockIdx.x >> 1;
    const int tid  = threadIdx.x;
    const int lane = tid & 31;
    const int wv   = tid >> 5;
    const int lo   = lane & 15;
    const int hi   = lane >> 4;

    const float* Ph = p + (b * S_) * (2 * H_) + hf * H_;
    const float* Qh = q + (b * S_) * (2 * H_) + hf * H_;

    v8f c0 = {}, c1 = {}, c2 = {}, c3 = {};
    float nacc = 0.f;

    for (int kt = 0; kt < H_; kt += KT) {
        __syncthreads();
        for (int i = tid; i < S_ * KT; i += 128) {
            int s = i / KT, k = i % KT;
            As[s][k] = Ph[s * (2 * H_) + kt + k];
            Bs[s][k] = Qh[s * (2 * H_) + kt + k];
        }
        __syncthreads();
        if (tid < S_) {
            for (int k = 0; k < KT; ++k) { float v = As[tid][k]; nacc += v * v; }
        } else {
            int r = tid - S_;
            for (int k = 0; k < KT; ++k) { float v = Bs[r][k]; nacc += v * v; }
        }
        const int m = wv * 16 + lo;
        for (int kb = 0; kb < KT; kb += 4) {
            int ka = kb + 2 * hi;
            v2f a  = *(const v2f*)&As[m][ka];
            v2f b0 = *(const v2f*)&Bs[ 0 + lo][ka];
            v2f b1 = *(const v2f*)&Bs[16 + lo][ka];
            v2f b2 = *(const v2f*)&Bs[32 + lo][ka];
            v2f b3 = *(const v2f*)&Bs[48 + lo][ka];
            c0 = __builtin_amdgcn_wmma_f32_16x16x4_f32(false, a, false, b0, (short)0, c0, false, false);
            c1 = __builtin_amdgcn_wmma_f32_16x16x4_f32(false, a, false, b1, (short)0, c1, false, false);
            c2 = __builtin_amdgcn_wmma_f32_16x16x4_f32(false, a, false, b2, (short)0, c2, false, false);
            c3 = __builtin_amdgcn_wmma_f32_16x16x4_f32(false, a, false, b3, (short)0, c3, false, false);
        }
    }
    __syncthreads();
    if (tid < S_) nA[tid] = sqrtf(nacc); else nB[tid - S_] = sqrtf(nacc);
    __syncthreads();

    #pragma unroll
    for (int j = 0; j < 8; ++j) {
        int m = wv * 16 + 8 * hi + j;
        float na = nA[m];
        cosBuf[m][ 0 + lo] = c0[j] / (na * nB[ 0 + lo]);
        cosBuf[m][16 + lo] = c1[j] / (na * nB[16 + lo]);
        cosBuf[m][32 + lo] = c2[j] / (na * nB[32 + lo]);
        cosBuf[m][48 + lo] = c3[j] / (na * nB[48 + lo]);
    }
    __syncthreads();

    float* dst = cosW + (b * 2 + hf) * (S_ * S_);
    for (int i = tid; i < S_ * S_; i += 128)
        dst[i] = cosBuf[i >> 6][i & 63];

    if (tid < S_) {                       // rowSum[s] = sum_q cos[s][q]
        float sm = 0.f;
        for (int n = 0; n < S_; ++n) sm += cosBuf[tid][n];
        rowSum[(b * 2 + hf) * S_ + tid] = sm;
    } else {                              // colSum[s] = sum_p cos[p][s]
        int n = tid - S_;
        float sm = 0.f;
        for (int m = 0; m < S_; ++m) sm += cosBuf[m][n];
        colSum[(b * 2 + hf) * S_ + n] = sm;
    }
}

// ---------------------------------------------------------------------------
// Kernel 3: attentive context vectors.  One block per (b, half, s).
//   q_vec_mean[s,h] = sum_j Q[j,h]*cos[s,j] / rowSum[s]
//   p_vec_mean[s,h] = sum_j P[j,h]*cos[j,s] / colSum[s]
//   q_vec_max[s,h]  = max_j Q[j,h]*cos[s,j];  p_vec_max analogous.
// ---------------------------------------------------------------------------
__global__ __launch_bounds__(128)
void attn_vec_kernel(const float* __restrict__ p, const float* __restrict__ q,
                     const float* __restrict__ cosW,
                     const float* __restrict__ rowSum, const float* __restrict__ colSum,
                     float* __restrict__ qvm, float* __restrict__ pvm,
                     float* __restrict__ qvx, float* __restrict__ pvx)
{
    __shared__ float cosRow[S_];
    __shared__ float cosCol[S_];

    const int bid = blockIdx.x;
    const int s   = bid % S_;
    const int hf  = (bid / S_) & 1;
    const int b   = bid / (2 * S_);
    const int tid = threadIdx.x;

    const float* cosBase = cosW + (b * 2 + hf) * (S_ * S_);
    if (tid < S_)       cosRow[tid]       = cosBase[s * S_ + tid];
    else                cosCol[tid - S_]  = cosBase[(tid - S_) * S_ + s];
    __syncthreads();

    const float rs = rowSum[(b * 2 + hf) * S_ + s];
    const float cs = colSum[(b * 2 + hf) * S_ + s];

    const float* Ph = p + (b * S_) * (2 * H_) + hf * H_;
    const float* Qh = q + (b * S_) * (2 * H_) + hf * H_;
    const int obase = ((b * 2 + hf) * S_ + s) * H_;

    for (int h = tid; h < H_; h += 128) {
        float qm = 0.f, pm = 0.f, qx = NEG_INF, px = NEG_INF;
        for (int j = 0; j < S_; ++j) {
            float a  = Qh[j * (2 * H_) + h] * cosRow[j];
            float bb = Ph[j * (2 * H_) + h] * cosCol[j];
            qm += a;  qx = fmaxf(qx, a);
            pm += bb; px = fmaxf(px, bb);
        }
        qvm[obase + h] = qm / rs;
        qvx[obase + h] = qx;
        pvm[obase + h] = pm / cs;
        pvx[obase + h] = px;
    }
}

// ---------------------------------------------------------------------------
// Kernel 4: weighted multi-perspective cosine (sections 0,1,4,5,6,7 for both
// mv_p and mv_q).  den = max(|xw|*|yw|, 1e-8) as in _mp_cos.
// ---------------------------------------------------------------------------
__global__ __launch_bounds__(256)
void mpcos_kernel(const float* __restrict__ p, const float* __restrict__ q,
                  const float* __restrict__ W,
                  const float* __restrict__ qvm, const float* __restrict__ pvm,
                  const float* __restrict__ qvx, const float* __restrict__ pvx,
                  float* __restrict__ out)
{
    const int TOT = 2 * 6 * B_ * S_ * L_;
    int idx = blockIdx.x * blockDim.x + threadIdx.x;
    if (idx >= TOT) return;

    const int l = idx % L_;  int t = idx / L_;
    const int s = t % S_;    t /= S_;
    const int b = t % B_;    t /= B_;
    const int sec = t % 6;   const int tgt = t / 6;   // 0 = mv_p, 1 = mv_q

    const int secTab[6] = {0, 1, 4, 5, 6, 7};
    const int secId = secTab[sec];
    const int hf = secId & 1;

    const float* X = (tgt == 0 ? p : q) + (b * S_ + s) * (2 * H_) + hf * H_;
    const float* Y;
    if (secId == 0)      Y = (tgt == 0 ? q : p) + (b * S_ + (S_ - 1)) * (2 * H_);
    else if (secId == 1) Y = (tgt == 0 ? q : p) + (b * S_) * (2 * H_) + H_;
    else {
        const float* vec = (secId <= 5) ? (tgt == 0 ? qvm : pvm)
                                        : (tgt == 0 ? qvx : pvx);
        Y = vec + ((b * 2 + hf) * S_ + s) * H_;
    }
    const float* Wr = W + (secId * L_ + l) * H_;

    float num = 0.f, nx = 0.f, ny = 0.f;
    for (int h = 0; h < H_; ++h) {
        float w  = Wr[h];
        float xw = X[h] * w;
        float yw = Y[h] * w;
        num += xw * yw;
        nx  += xw * xw;
        ny  += yw * yw;
    }
    float den = fmaxf(sqrtf(nx) * sqrtf(ny), 1e-8f);
    out[(tgt == 0 ? 0 : OUTQOFF) + (b * S_ + s) * OUTC + secId * L_ + l] = num / den;
}

// ---------------------------------------------------------------------------
extern "C" void kernel_launch(void* const* d_in, const int* in_sizes, int n_in,
                              void* d_out, int out_size, void* d_ws, size_t ws_size,
                              hipStream_t stream)
{
    const float* p = (const float*)d_in[0];   // (32, 64, 600)
    const float* q = (const float*)d_in[1];   // (32, 64, 600)
    const float* W = (const float*)d_in[2];   // (8, 20, 300)
    float* out = (float*)d_out;               // mv_p (32,64,160) then mv_q

    float* ws     = (float*)d_ws;
    float* cosW   = ws;                              // 32*2*64*64   = 262144
    float* rowSum = cosW + B_ * 2 * S_ * S_;         // 32*2*64      = 4096
    float* colSum = rowSum + B_ * 2 * S_;            // 4096
    float* qvm    = colSum + B_ * 2 * S_;            // each 32*2*64*300 = 1228800
    float* pvm    = qvm + B_ * 2 * S_ * H_;
    float* qvx    = pvm + B_ * 2 * S_ * H_;
    float* pvx    = qvx + B_ * 2 * S_ * H_;

    maxpool_kernel <<<B_ * 2 * L_, 128, 0, stream>>>(p, q, W, out);
    attn_cos_kernel<<<B_ * 2,      128, 0, stream>>>(p, q, cosW, rowSum, colSum);
    attn_vec_kernel<<<B_ * 2 * S_, 128, 0, stream>>>(p, q, cosW, rowSum, colSum,
                                                     qvm, pvm, qvx, pvx);
    const int TOT = 2 * 6 * B_ * S_ * L_;
    mpcos_kernel   <<<(TOT + 255) / 256, 256, 0, stream>>>(p, q, W, qvm, pvm, qvx, pvx, out);
}